// MergeNN_22153441313294
// MI455X (gfx1250) — compile-verified
//
#include <hip/hip_runtime.h>
#include <stdint.h>

#define KK  2048   // K rows of x / outputs
#define NN  8192   // N star / dataset rows
#define DD  1024   // feature dim
#define HH  1024   // hidden dim
#define DYY 10     // label dim
#define CC  10     // classes

typedef __attribute__((ext_vector_type(2))) float v2f;
typedef __attribute__((ext_vector_type(8))) float v8f;

// ---------------------------------------------------------------------------
// Register-blocked fp32 WMMA: one wave computes an (MT*16) x (NT*16) block.
// BT=true  : B row-major [cols, kdim] (A @ B^T)  -> contiguous v2f loads
// BT=false : B row-major [kdim, cols] (A @ B)    -> per-lane b32 loads,
//            coalesced across the wave (16 consecutive floats per B row).
// A fragment layout per ISA: lanes 0-15 rows M, K=(k,k+1); lanes 16-31 same
// rows, K=(k+2,k+3). B mirrored over columns. C/D: VGPR v -> row v (+8 for
// lanes 16-31), col = lane&15.
// ---------------------------------------------------------------------------
template <int MT, int NT, bool BT>
__device__ inline void gemm_block(const float* __restrict__ A, int lda,
                                  const float* __restrict__ B, int ldb,
                                  int row0, int col0, int kdim,
                                  v8f acc[MT][NT])
{
  const int lane = threadIdx.x & 31;
  const int half = lane >> 4;   // 0: K pair (k,k+1)   1: K pair (k+2,k+3)
  const int l16  = lane & 15;

  const float* arow[MT];
#pragma unroll
  for (int mi = 0; mi < MT; ++mi)
    arow[mi] = A + (size_t)(row0 + mi * 16 + l16) * lda;
  const float* brow[NT];
#pragma unroll
  for (int ni = 0; ni < NT; ++ni)
    brow[ni] = BT ? (B + (size_t)(col0 + ni * 16 + l16) * ldb) : B;

  for (int k = 0; k < kdim; k += 4) {
    const int ka = k + 2 * half;
    v2f a[MT], b[NT];
#pragma unroll
    for (int mi = 0; mi < MT; ++mi) a[mi] = *(const v2f*)(arow[mi] + ka);
#pragma unroll
    for (int ni = 0; ni < NT; ++ni) {
      if (BT) {
        b[ni] = *(const v2f*)(brow[ni] + ka);
      } else {
        b[ni].x = B[(size_t)ka       * ldb + col0 + ni * 16 + l16];
        b[ni].y = B[(size_t)(ka + 1) * ldb + col0 + ni * 16 + l16];
      }
    }
#pragma unroll
    for (int mi = 0; mi < MT; ++mi)
#pragma unroll
      for (int ni = 0; ni < NT; ++ni)
        acc[mi][ni] = __builtin_amdgcn_wmma_f32_16x16x4_f32(
            false, a[mi], false, b[ni], (short)0, acc[mi][ni], false, false);
  }
}

__device__ inline float wave_reduce_sum(float v) {
  for (int m = 16; m >= 1; m >>= 1) v += __shfl_xor(v, m, 32);
  return v;
}

// ---------------------------------------------------------------------------
// Init accumulators (d_ws is poisoned; must re-init every call).
// ---------------------------------------------------------------------------
__global__ void k_init(unsigned long long* minpack, float* denom,
                       float* num1, float* den1, float* num2, float* den2)
{
  int i = blockIdx.x * blockDim.x + threadIdx.x;
  if (i < KK) {
    minpack[i] = 0xFFFFFFFFFFFFFFFFull;
    denom[i] = 0.f; den1[i] = 0.f; den2[i] = 0.f;
  }
  if (i < KK * DYY) { num1[i] = 0.f; num2[i] = 0.f; }
}

// ---------------------------------------------------------------------------
// Row squared norms: one wave per row.
// ---------------------------------------------------------------------------
__global__ void k_rownorm(const float* __restrict__ M, float* __restrict__ out,
                          int rows, int cols)
{
  int wid  = (blockIdx.x * blockDim.x + threadIdx.x) >> 5;
  int lane = threadIdx.x & 31;
  if (wid >= rows) return;
  const float* r = M + (size_t)wid * cols;
  float s = 0.f;
  for (int j = lane; j < cols; j += 32) { float t = r[j]; s += t * t; }
  s = wave_reduce_sum(s);
  if (lane == 0) out[wid] = s;
}

// ---------------------------------------------------------------------------
// GEMM 1: S = star_feats @ x^T ; epilogue d2 = max(a2+b2-2S,0), store exp(-d2)
// transposed into expsT[K,N]; per-k-column min/argmin/denominator via atomics.
// rows = n (star, 2 tiles), cols = k (x, 4 tiles). 8192 waves.
// ---------------------------------------------------------------------------
__global__ void k_gemm_exps(const float* __restrict__ sf, const float* __restrict__ x,
                            const float* __restrict__ sf2, const float* __restrict__ x2,
                            float* __restrict__ expsT,
                            unsigned long long* __restrict__ minpack,
                            float* __restrict__ denom)
{
  const int gw = (blockIdx.x * blockDim.x + threadIdx.x) >> 5;
  const int cb = gw & (KK / 64 - 1);   // col block (32)
  const int rb = gw >> 5;              // row block (256)
  const int n0 = rb * 32, k0 = cb * 64;

  v8f acc[2][4] = {};
  gemm_block<2, 4, true>(sf, DD, x, DD, n0, k0, DD, acc);

  const int lane = threadIdx.x & 31;
  const int half = lane >> 4;
  const int l16  = lane & 15;

  float sn2[16];
#pragma unroll
  for (int mi = 0; mi < 2; ++mi)
#pragma unroll
    for (int v = 0; v < 8; ++v)
      sn2[mi * 8 + v] = sf2[n0 + mi * 16 + v + 8 * half];

#pragma unroll
  for (int ni = 0; ni < 4; ++ni) {
    const int kcol = k0 + ni * 16 + l16;
    const float bk = x2[kcol];
    float minv = 3.4e38f; int minn = 0; float ssum = 0.f;
#pragma unroll
    for (int mi = 0; mi < 2; ++mi) {
#pragma unroll
      for (int v = 0; v < 8; ++v) {
        int nrow = n0 + mi * 16 + v + 8 * half;
        float d2 = fmaxf(sn2[mi * 8 + v] + bk - 2.f * acc[mi][ni][v], 0.f);
        float e  = __expf(-d2);
        expsT[(size_t)kcol * NN + nrow] = e;
        ssum += e;
        if (d2 < minv || (d2 == minv && nrow < minn)) { minv = d2; minn = nrow; }
      }
    }
    // combine the two halves (same k column lives in lane and lane^16)
    float ominv = __shfl_xor(minv, 16, 32);
    int   ominn = __shfl_xor(minn, 16, 32);
    float osum  = __shfl_xor(ssum, 16, 32);
    ssum += osum;
    if (ominv < minv || (ominv == minv && ominn < minn)) { minv = ominv; minn = ominn; }
    if (half == 0) {
      unsigned long long pack =
          ((unsigned long long)__float_as_uint(minv) << 32) | (unsigned)minn;
      atomicMin(minpack + kcol, pack);
      atomicAdd(denom + kcol, ssum);
    }
  }
}

// ---------------------------------------------------------------------------
// GEMM 2: xt = (expsT[K,N] @ feats[N,D]) / denom, exact-match override.
// rows = k (2 tiles), cols = d (4 tiles). 1024 waves.
// ---------------------------------------------------------------------------
__global__ void k_transport(const float* __restrict__ expsT,
                            const float* __restrict__ feats,
                            const float* __restrict__ denom,
                            const unsigned long long* __restrict__ minpack,
                            float* __restrict__ xt)
{
  const int gw = (blockIdx.x * blockDim.x + threadIdx.x) >> 5;
  const int cb = gw & (DD / 64 - 1);   // col block (16)
  const int rb = gw >> 4;              // row block (64)
  const int k0 = rb * 32, d0 = cb * 64;

  v8f acc[2][4] = {};
  gemm_block<2, 4, false>(expsT, NN, feats, DD, k0, d0, NN, acc);

  const int lane = threadIdx.x & 31;
  const int half = lane >> 4;
  const int l16  = lane & 15;
#pragma unroll
  for (int mi = 0; mi < 2; ++mi) {
#pragma unroll
    for (int v = 0; v < 8; ++v) {
      const int row = k0 + mi * 16 + v + 8 * half;
      const unsigned long long mp = minpack[row];
      const float md2 = __uint_as_float((unsigned)(mp >> 32));
      const unsigned midx = (unsigned)mp;
      const bool matched = (md2 < 1e-9f);
      const float inv_den = 1.f / denom[row];
#pragma unroll
      for (int ni = 0; ni < 4; ++ni) {
        const int col = d0 + ni * 16 + l16;
        float val = matched ? feats[(size_t)midx * DD + col]
                            : acc[mi][ni][v] * inv_den;
        xt[(size_t)row * DD + col] = val;
      }
    }
  }
}

// ---------------------------------------------------------------------------
// GEMM 3: h = relu(xt @ Wa + ba)   [K,H]. 1024 waves.
// ---------------------------------------------------------------------------
__global__ void k_mlp_hidden(const float* __restrict__ xt,
                             const float* __restrict__ Wa,
                             const float* __restrict__ ba,
                             float* __restrict__ h)
{
  const int gw = (blockIdx.x * blockDim.x + threadIdx.x) >> 5;
  const int cb = gw & (HH / 64 - 1);   // col block (16)
  const int rb = gw >> 4;              // row block (64)
  const int k0 = rb * 32, h0 = cb * 64;

  v8f acc[2][4] = {};
  gemm_block<2, 4, false>(xt, DD, Wa, HH, k0, h0, DD, acc);

  const int lane = threadIdx.x & 31;
  const int half = lane >> 4;
  const int l16  = lane & 15;
  float bias[4];
#pragma unroll
  for (int ni = 0; ni < 4; ++ni) bias[ni] = ba[h0 + ni * 16 + l16];
#pragma unroll
  for (int mi = 0; mi < 2; ++mi)
#pragma unroll
    for (int v = 0; v < 8; ++v) {
      const int row = k0 + mi * 16 + v + 8 * half;
#pragma unroll
      for (int ni = 0; ni < 4; ++ni) {
        const int col = h0 + ni * 16 + l16;
        h[(size_t)row * HH + col] = fmaxf(acc[mi][ni][v] + bias[ni], 0.f);
      }
    }
}

// ---------------------------------------------------------------------------
// y = h @ Wb + bb (DY=10, tiny) fused with class argmin. One wave per k row.
// ---------------------------------------------------------------------------
__global__ void k_mlp2cls(const float* __restrict__ h, const float* __restrict__ Wb,
                          const float* __restrict__ bb, const float* __restrict__ uniq,
                          int* __restrict__ cls_out)
{
  int wid  = (blockIdx.x * blockDim.x + threadIdx.x) >> 5;
  int lane = threadIdx.x & 31;
  if (wid >= KK) return;
  const float* hr = h + (size_t)wid * HH;
  float acc[DYY];
#pragma unroll
  for (int dy = 0; dy < DYY; ++dy) acc[dy] = 0.f;
  for (int j = lane; j < HH; j += 32) {
    float hv = hr[j];
    const float* wr = Wb + (size_t)j * DYY;
#pragma unroll
    for (int dy = 0; dy < DYY; ++dy) acc[dy] += hv * wr[dy];
  }
#pragma unroll
  for (int dy = 0; dy < DYY; ++dy) acc[dy] = wave_reduce_sum(acc[dy]);
  if (lane == 0) {
    float best = 3.4e38f; int bc = 0;
    for (int c = 0; c < CC; ++c) {
      float d = 0.f;
#pragma unroll
      for (int dy = 0; dy < DYY; ++dy) {
        float t = acc[dy] + bb[dy] - uniq[c * DYY + dy];
        d += t * t;
      }
      if (d < best) { best = d; bc = c; }
    }
    cls_out[wid] = bc;
  }
}

// ---------------------------------------------------------------------------
// GEMM 4: d2 = sqdist(xt, feats), w = exp(-d2)*[c_k==cls_n];
// accumulate w@star_labels and w-sums. The 4 column-tiles share output rows,
// so contributions are combined in registers before shuffle-reduce + atomics.
// rows = k (2 tiles), cols = n (4 tiles). 8192 waves.
// ---------------------------------------------------------------------------
__global__ void k_tback(const float* __restrict__ xt, const float* __restrict__ feats,
                        const float* __restrict__ xtn2, const float* __restrict__ fn2,
                        const int* __restrict__ cpred, const int* __restrict__ cfrom,
                        const float* __restrict__ slab,
                        float* __restrict__ num, float* __restrict__ den)
{
  const int gw = (blockIdx.x * blockDim.x + threadIdx.x) >> 5;
  const int cb = gw & (NN / 64 - 1);   // col block (128)
  const int rb = gw >> 7;              // row block (64)
  const int k0 = rb * 32, n0 = cb * 64;

  v8f acc[2][4] = {};
  gemm_block<2, 4, true>(xt, DD, feats, DD, k0, n0, DD, acc);

  const int lane = threadIdx.x & 31;
  const int half = lane >> 4;
  const int l16  = lane & 15;

  float f2a[4]; int cna[4]; float lab[4][DYY];
#pragma unroll
  for (int ni = 0; ni < 4; ++ni) {
    const int ncol = n0 + ni * 16 + l16;
    f2a[ni] = fn2[ncol];
    cna[ni] = cfrom[ncol];
    const float* lr = slab + (size_t)ncol * DYY;
#pragma unroll
    for (int dy = 0; dy < DYY; ++dy) lab[ni][dy] = lr[dy];
  }

#pragma unroll
  for (int mi = 0; mi < 2; ++mi) {
#pragma unroll
    for (int v = 0; v < 8; ++v) {
      const int row = k0 + mi * 16 + v + 8 * half;
      const float xn = xtn2[row];
      const int cp = cpred[row];
      float wsum = 0.f;
      float p[DYY];
#pragma unroll
      for (int dy = 0; dy < DYY; ++dy) p[dy] = 0.f;
#pragma unroll
      for (int ni = 0; ni < 4; ++ni) {
        float d2 = fmaxf(xn + f2a[ni] - 2.f * acc[mi][ni][v], 0.f);
        float w = (cp == cna[ni]) ? __expf(-d2) : 0.f;
        wsum += w;
#pragma unroll
        for (int dy = 0; dy < DYY; ++dy) p[dy] += w * lab[ni][dy];
      }
      // reduce across the 16 columns of this half (xor masks stay within a half)
      for (int m = 8; m >= 1; m >>= 1) {
        wsum += __shfl_xor(wsum, m, 32);
#pragma unroll
        for (int dy = 0; dy < DYY; ++dy) p[dy] += __shfl_xor(p[dy], m, 32);
      }
      if (l16 == 0) {
        atomicAdd(den + row, wsum);
#pragma unroll
        for (int dy = 0; dy < DYY; ++dy)
          atomicAdd(num + (size_t)row * DYY + dy, p[dy]);
      }
    }
  }
}

__global__ void k_final(const float* __restrict__ num1, const float* __restrict__ den1,
                        const float* __restrict__ num2, const float* __restrict__ den2,
                        float* __restrict__ out)
{
  int i = blockIdx.x * blockDim.x + threadIdx.x;
  if (i < KK * DYY) {
    int k = i / DYY;
    out[i] = 0.5f * (num1[i] / den1[k] + num2[i] / den2[k]);
  }
}

// ---------------------------------------------------------------------------
extern "C" void kernel_launch(void* const* d_in, const int* in_sizes, int n_in,
                              void* d_out, int out_size, void* d_ws, size_t ws_size,
                              hipStream_t stream)
{
  const float* x    = (const float*)d_in[0];
  const float* sf   = (const float*)d_in[1];
  const float* slab = (const float*)d_in[2];
  const float* f1   = (const float*)d_in[3];
  const float* f2   = (const float*)d_in[4];
  const float* u1   = (const float*)d_in[5];
  const float* u2   = (const float*)d_in[6];
  const float* W1a  = (const float*)d_in[7];
  const float* b1a  = (const float*)d_in[8];
  const float* W1b  = (const float*)d_in[9];
  const float* b1b  = (const float*)d_in[10];
  const float* W2a  = (const float*)d_in[11];
  const float* b2a  = (const float*)d_in[12];
  const float* W2b  = (const float*)d_in[13];
  const float* b2b  = (const float*)d_in[14];
  const int*   cls1 = (const int*)d_in[15];
  const int*   cls2 = (const int*)d_in[16];
  float* out = (float*)d_out;

  // ---- workspace layout (~80.4 MB) ----
  char* base = (char*)d_ws;
  float* expsT = (float*)base;                       // K*N fp32 (64 MB), dead after transports
  float* h1 = expsT;                                 // alias: K*H (8 MB)
  float* h2 = expsT + (size_t)KK * HH;               // alias: K*H (8 MB)
  float* xt1 = (float*)(base + (size_t)KK * NN * 4); // 8 MB
  float* xt2 = xt1 + (size_t)KK * DD;                // 8 MB
  float* sm = xt2 + (size_t)KK * DD;
  float* sf2   = sm; sm += NN;
  float* x2n   = sm; sm += KK;
  float* f1n2  = sm; sm += NN;
  float* f2n2  = sm; sm += NN;
  float* xt1n2 = sm; sm += KK;
  float* xt2n2 = sm; sm += KK;
  float* denom = sm; sm += KK;
  float* num1  = sm; sm += KK * DYY;
  float* den1  = sm; sm += KK;
  float* num2  = sm; sm += KK * DYY;
  float* den2  = sm; sm += KK;
  unsigned long long* minpack = (unsigned long long*)sm;  // 8-aligned (all counts even)
  int* c1 = (int*)(minpack + KK);
  int* c2 = c1 + KK;
  (void)in_sizes; (void)n_in; (void)out_size; (void)ws_size;

  const dim3 B(256);  // 8 wave32 per block

  k_init<<<(KK * DYY + 255) / 256, B, 0, stream>>>(minpack, denom, num1, den1, num2, den2);

  k_rownorm<<<NN / 8, B, 0, stream>>>(sf, sf2, NN, DD);
  k_rownorm<<<KK / 8, B, 0, stream>>>(x, x2n, KK, DD);
  k_rownorm<<<NN / 8, B, 0, stream>>>(f1, f1n2, NN, DD);
  k_rownorm<<<NN / 8, B, 0, stream>>>(f2, f2n2, NN, DD);

  // kernel matrix + exact-match stats: (N/32)*(K/64) = 8192 waves = 1024 blocks
  k_gemm_exps<<<(NN / 32) * (KK / 64) / 8, B, 0, stream>>>(sf, x, sf2, x2n,
                                                           expsT, minpack, denom);

  // barycentric transport: (K/32)*(D/64) = 1024 waves = 128 blocks
  k_transport<<<(KK / 32) * (DD / 64) / 8, B, 0, stream>>>(expsT, f1, denom, minpack, xt1);
  k_transport<<<(KK / 32) * (DD / 64) / 8, B, 0, stream>>>(expsT, f2, denom, minpack, xt2);

  k_rownorm<<<KK / 8, B, 0, stream>>>(xt1, xt1n2, KK, DD);
  k_rownorm<<<KK / 8, B, 0, stream>>>(xt2, xt2n2, KK, DD);

  // MLP hidden layers (h1/h2 alias the now-dead expsT region)
  k_mlp_hidden<<<(KK / 32) * (HH / 64) / 8, B, 0, stream>>>(xt1, W1a, b1a, h1);
  k_mlp_hidden<<<(KK / 32) * (HH / 64) / 8, B, 0, stream>>>(xt2, W2a, b2a, h2);

  k_mlp2cls<<<KK / 8, B, 0, stream>>>(h1, W1b, b1b, u1, c1);
  k_mlp2cls<<<KK / 8, B, 0, stream>>>(h2, W2b, b2b, u2, c2);

  // class-restricted Gaussian label transport: (K/32)*(N/64) = 8192 waves
  k_tback<<<(KK / 32) * (NN / 64) / 8, B, 0, stream>>>(xt1, f1, xt1n2, f1n2,
                                                       c1, cls1, slab, num1, den1);
  k_tback<<<(KK / 32) * (NN / 64) / 8, B, 0, stream>>>(xt2, f2, xt2n2, f2n2,
                                                       c2, cls2, slab, num2, den2);

  k_final<<<(KK * DYY + 255) / 256, B, 0, stream>>>(num1, den1, num2, den2, out);
}